// DynamicConv3d_71691594104906
// MI455X (gfx1250) — compile-verified
//
#include <hip/hip_runtime.h>
#include <hip/hip_bf16.h>

typedef __attribute__((ext_vector_type(2))) float v2f;
typedef __attribute__((ext_vector_type(8))) float v8f;

#define N_S 8
#define C_IN 64
#define T_D 30
#define H_D 64
#define W_D 44
#define O_CH 128
#define KTAPS 27
#define KTOT (C_IN * KTAPS)      // 1728
#define HW (H_D * W_D)           // 2816
#define SPATIAL (T_D * HW)       // 84480
#define GH 32
#define NE 4

#define NT 8                     // N-subtiles of 16 per wave
#define COLS (NT * 16)           // 128 output columns per WG (84480/128 = 660)
#define KB 64                    // K-steps staged per LDS block
#define NKB (KTOT / KB)          // 27
#define PSTR (COLS * 2 + 2)      // K-pair row stride (floats), pad keeps b64 aligned
#define BUFSZ (32 * PSTR)        // one im2col buffer: 32 K-pairs x PSTR
#define SMEM_BYTES ((2 * BUFSZ + 32) * 4)

// ---------------- Stage 1: global average pool  context[n][c] ----------------
__global__ __launch_bounds__(256) void pool_kernel(const float* __restrict__ x,
                                                   float* __restrict__ context) {
    const int c = blockIdx.x, n = blockIdx.y;
    const float* p = x + (size_t)(n * C_IN + c) * SPATIAL;
    float s = 0.f;
    for (int i = threadIdx.x; i < SPATIAL; i += 256) s += p[i];
    __shared__ float red[256];
    red[threadIdx.x] = s;
    __syncthreads();
    for (int off = 128; off > 0; off >>= 1) {
        if (threadIdx.x < off) red[threadIdx.x] += red[threadIdx.x + off];
        __syncthreads();
    }
    if (threadIdx.x == 0) context[n * C_IN + c] = red[0] * (1.0f / SPATIAL);
}

// ---------------- Stage 2: gate MLP + softmax  gate[n][e] ----------------
__global__ __launch_bounds__(256) void gate_kernel(const float* __restrict__ context,
                                                   const float* __restrict__ w1,
                                                   const float* __restrict__ b1,
                                                   const float* __restrict__ w2,
                                                   const float* __restrict__ b2,
                                                   float* __restrict__ gate) {
    __shared__ float hid[N_S][GH];
    __shared__ float logits[N_S][NE];
    const int tid = threadIdx.x;
    const int n = tid >> 5, g = tid & 31;   // 8 * 32 = 256 threads
    float a = b1[g];
    for (int c = 0; c < C_IN; ++c) a += context[n * C_IN + c] * w1[g * C_IN + c];
    hid[n][g] = fmaxf(a, 0.f);
    __syncthreads();
    if (tid < N_S * NE) {
        const int nn = tid >> 2, e = tid & 3;
        float l = b2[e];
        for (int g2 = 0; g2 < GH; ++g2) l += hid[nn][g2] * w2[e * GH + g2];
        logits[nn][e] = l;
    }
    __syncthreads();
    if (tid < N_S) {
        float mx = logits[tid][0];
        for (int e = 1; e < NE; ++e) mx = fmaxf(mx, logits[tid][e]);
        float s = 0.f, ex[NE];
        for (int e = 0; e < NE; ++e) { ex[e] = __expf(logits[tid][e] - mx); s += ex[e]; }
        const float inv = 1.0f / s;
        for (int e = 0; e < NE; ++e) gate[tid * NE + e] = ex[e] * inv;
    }
}

// ---------------- Stage 3: per-sample mixed kernels  Kmix[n][o][i] ----------------
__global__ __launch_bounds__(256) void mix_kernel(const float* __restrict__ gate,
                                                  const float* __restrict__ weight,
                                                  float* __restrict__ kmix) {
    const int idx = blockIdx.x * 256 + threadIdx.x;      // exactly covers 8*128*1728
    const int per = O_CH * KTOT;
    const int n = idx / per;
    const int rem = idx - n * per;
    float s = 0.f;
    #pragma unroll
    for (int e = 0; e < NE; ++e)
        s += gate[n * NE + e] * weight[(size_t)e * per + rem];
    kmix[idx] = s;
}

// ---------------- Stage 4: WMMA f32 grouped conv3d as GEMM ----------------
// WG = sample n x 128 output columns; wave w owns O-tile w (16 rows) x 8 N-subtiles.
// Double-buffered LDS im2col in K-pair-interleaved layout: fragment = 1 ds_load_b64.
__global__ __launch_bounds__(256) void conv_wmma_kernel(const float* __restrict__ x,
                                                        const float* __restrict__ kmix,
                                                        float* __restrict__ y) {
    extern __shared__ float smem[];              // 2 * BUFSZ floats + tap-offset table
    float* xc0 = smem;
    float* xc1 = smem + BUFSZ;
    int* doffs = (int*)(smem + 2 * BUFSZ);       // 27 tap offsets

    const int tid = threadIdx.x;
    const int n = blockIdx.y;
    const int pbase = blockIdx.x * COLS;

    // ---- fill role: fixed column; K rows frow0, frow0+2, ... (stride 2) ----
    const int fcol = tid & (COLS - 1);
    const int frow0 = tid >> 7;                  // 0..1
    const int p = pbase + fcol;
    const int t = p / HW;
    const int rhw = p - t * HW;
    const int h = rhw / W_D;
    const int w = rhw - h * W_D;
    const float* xn = x + (size_t)n * C_IN * SPATIAL;
    const long boff = (long)(t - 1) * HW + (long)(h - 1) * W_D + (w - 1);

    if (tid < KTAPS) {                            // tap offset table (once)
        const int dt = tid / 9, r9 = tid - dt * 9;
        doffs[tid] = dt * HW + (r9 / 3) * W_D + (r9 % 3);
    }
    // 27-bit boundary validity mask for this column (once)
    unsigned mask = 0;
    {
        int dd = 0;
        #pragma unroll
        for (int dt = 0; dt < 3; ++dt)
            #pragma unroll
            for (int dh = 0; dh < 3; ++dh)
                #pragma unroll
                for (int dw = 0; dw < 3; ++dw, ++dd) {
                    const bool ok = (unsigned)(t + dt - 1) < (unsigned)T_D &&
                                    (unsigned)(h + dh - 1) < (unsigned)H_D &&
                                    (unsigned)(w + dw - 1) < (unsigned)W_D;
                    mask |= (ok ? 1u : 0u) << dd;
                }
    }
    int fc = 0, fd = frow0;                      // running (channel, tap) for K index

    // ---- compute role: WMMA fragment ownership (wave32, ISA 16x4 A layout) ----
    const int lane = tid & 31;
    const int wv = tid >> 5;                     // O-tile 0..7
    const int m = lane & 15;
    const int hi = lane >> 4;                    // lanes 16-31 hold K+2
    const float* arow = kmix + ((size_t)n * O_CH + (wv * 16 + m)) * KTOT;

    v8f acc[NT] = {};

    auto fill = [&](float* dst) {
        #pragma unroll
        for (int i = 0; i < 32; ++i) {           // K rows frow0 + 2*i
            const int off = doffs[fd];
            float v = 0.f;
            if ((mask >> fd) & 1u)
                v = xn[boff + (long)fc * SPATIAL + off];
            dst[i * PSTR + (fcol << 1) + frow0] = v;   // kpair-interleaved
            fd += 2;
            if (fd >= KTAPS) { fd -= KTAPS; ++fc; }
        }
    };
    auto compute = [&](const float* bsrc, int kbase) {
        #pragma unroll
        for (int kk = 0; kk < 16; ++kk) {
            const v2f a = *(const v2f*)(arow + kbase + kk * 4 + 2 * hi);
            const float* br = bsrc + (2 * kk + hi) * PSTR + (m << 1);
            #pragma unroll
            for (int s = 0; s < NT; ++s) {
                const v2f b = *(const v2f*)(br + 32 * s);   // B[K=2hi..2hi+1][N]
                acc[s] = __builtin_amdgcn_wmma_f32_16x16x4_f32(
                    false, a, false, b, (short)0, acc[s], false, false);
            }
        }
    };

    __syncthreads();                 // tap table ready
    fill(xc0);                       // stage K-block 0
    __syncthreads();

    for (int kb = 0; kb < NKB; ++kb) {
        if (kb + 1 < NKB) {
            __builtin_prefetch(arow + (kb + 1) * KB, 0, 3);
            fill((kb & 1) ? xc0 : xc1);          // stage next while computing
        }
        compute((kb & 1) ? xc1 : xc0, kb * KB);  // 128 WMMAs per wave per barrier
        __syncthreads();
    }

    // D layout: VGPR r -> M = r + 8*hi, N = lane&15 ; coalesced 16-float runs
    float* yn = y + (size_t)n * O_CH * SPATIAL;
    #pragma unroll
    for (int s = 0; s < NT; ++s) {
        const int col = pbase + s * 16 + m;
        #pragma unroll
        for (int r = 0; r < 8; ++r) {
            const int row = wv * 16 + r + 8 * hi;
            yn[(size_t)row * SPATIAL + col] = acc[s][r];
        }
    }
}

extern "C" void kernel_launch(void* const* d_in, const int* in_sizes, int n_in,
                              void* d_out, int out_size, void* d_ws, size_t ws_size,
                              hipStream_t stream) {
    const float* x      = (const float*)d_in[0];
    const float* weight = (const float*)d_in[1];
    const float* gw1    = (const float*)d_in[2];
    const float* gb1    = (const float*)d_in[3];
    const float* gw2    = (const float*)d_in[4];
    const float* gb2    = (const float*)d_in[5];
    float* out = (float*)d_out;

    float* ws      = (float*)d_ws;
    float* kmix    = ws;                                   // 8*128*1728 floats (~7 MB)
    float* context = ws + (size_t)N_S * O_CH * KTOT;       // 512 floats
    float* gate    = context + N_S * C_IN;                 // 32 floats

    pool_kernel<<<dim3(C_IN, N_S), 256, 0, stream>>>(x, context);
    gate_kernel<<<1, 256, 0, stream>>>(context, gw1, gb1, gw2, gb2, gate);
    mix_kernel<<<(N_S * O_CH * KTOT) / 256, 256, 0, stream>>>(gate, weight, kmix);
    conv_wmma_kernel<<<dim3(SPATIAL / COLS, N_S), 256, SMEM_BYTES, stream>>>(x, kmix, out);
}